// SetTransformer_6305011991154
// MI455X (gfx1250) — compile-verified
//
#include <hip/hip_runtime.h>
#include <math.h>

#define B_GRAPHS 256
#define D 256
#define H 4
#define SEEDS 4
#define DH 64
#define NC 16            // H*SEEDS score combos, c = h*4 + s

typedef float v2f __attribute__((ext_vector_type(2)));
typedef float v8f __attribute__((ext_vector_type(8)));

// -------- workspace layout (float offsets from d_ws) --------
#define WS_OFFSETS 0          // 257 ints, occupies first 512 "floats"
#define WS_SCOREW 512         // 256*16 floats
#define WS_SBIAS  (512 + 4096)
#define WS_SCORES 8192        // N*16 floats  (~16 MB)

// -------- LDS layout (float offsets inside dynamic smem) --------
// s_sw     [0, 4096)              scoreW
// s_x      [4096, 4096+8*4160)    8 wave slots, 16 rows x stride 260
// s_red    [37376, 37504)         8x16 reduction scratch
// reuse of s_x region after phase 1:
//   s_Y    [4096, 8192)   16x256
//   s_ctx  [8192, 9216)
//   s_att  [9216, 10240)
//   s_h    [10240, 11264)
#define LDS_FLOATS 37504
#define XROW 260
#define XSLOT (16 * XROW)

// ============ Kernel 0: offsets scan + fold q into scoreW ============
__global__ __launch_bounds__(256)
void setx_prep(const int* __restrict__ counts,
               const float* __restrict__ seeds,
               const float* __restrict__ in_proj_w,
               const float* __restrict__ in_proj_b,
               float* __restrict__ wsf, int* __restrict__ wso)
{
    __shared__ float q[SEEDS * D];
    const int t = threadIdx.x;

    if (t == 0) {
        int acc = 0;
        for (int i = 0; i < B_GRAPHS; ++i) { wso[i] = acc; acc += counts[i]; }
        wso[B_GRAPHS] = acc;
    }

    // q[s][i] = seeds[s]·Wq_row(i) + bq[i]   (Wq = rows [0,256) of in_proj_w)
    for (int s = 0; s < SEEDS; ++s) {
        const float* wrow = in_proj_w + (size_t)t * D;
        const float* srow = seeds + s * D;
        float acc = in_proj_b[t];
        for (int j = 0; j < D; ++j) acc += srow[j] * wrow[j];
        q[s * D + t] = acc;
    }
    __syncthreads();

    // scoreW[j][c] = scale * sum_d Wk[(h*64+d), j] * q[s][h*64+d]
    {
        const int j = t;
        for (int c = 0; c < NC; ++c) {
            const int h = c >> 2, s = c & 3;
            float acc = 0.f;
            for (int d = 0; d < DH; ++d)
                acc += in_proj_w[(size_t)(D + h * DH + d) * D + j] * q[s * D + h * DH + d];
            wsf[WS_SCOREW + j * NC + c] = acc * 0.125f;   // 1/sqrt(64)
        }
    }
    if (t < NC) {
        const int h = t >> 2, s = t & 3;
        float acc = 0.f;
        for (int d = 0; d < DH; ++d)
            acc += in_proj_b[D + h * DH + d] * q[s * D + h * DH + d];
        wsf[WS_SBIAS + t] = acc * 0.125f;
    }
}

// ============ Kernel 1: per-graph scores + softmax + pooled output ============
__global__ __launch_bounds__(256)
void setx_main(const float* __restrict__ x,
               const int* __restrict__ wso,
               const float* __restrict__ wsf,
               float* __restrict__ scores,
               const float* __restrict__ in_proj_w,
               const float* __restrict__ in_proj_b,
               const float* __restrict__ out_proj_w,
               const float* __restrict__ out_proj_b,
               const float* __restrict__ fc1_w,
               const float* __restrict__ fc1_b,
               const float* __restrict__ fc2_w,
               const float* __restrict__ fc2_b,
               float* __restrict__ out)
{
    extern __shared__ float smem[];
    float* s_sw  = smem;
    float* s_x   = smem + 4096;
    float* s_red = smem + 37376;
    float* s_Y   = smem + 4096;
    float* s_ctx = smem + 8192;
    float* s_att = smem + 9216;
    float* s_h   = smem + 10240;

    const int b    = blockIdx.x;
    const int t    = threadIdx.x;
    const int wid  = t >> 5;
    const int lane = t & 31;
    const int l15  = lane & 15;
    const int kh   = lane >> 4;          // 0 / 1 (K half per WMMA layout)

    const int off    = wso[b];
    const int count  = wso[b + 1] - off;
    const int ntiles = (count + 15) >> 4;

    for (int i = t; i < 4096; i += 256) s_sw[i] = wsf[WS_SCOREW + i];
    const float sbias = wsf[WS_SBIAS + l15];
    __syncthreads();

    // ---------- Phase 1: scores = x @ scoreW (WMMA f32 16x16x4), track max ----------
    float* myx = s_x + wid * XSLOT;
    float wmaxv = -3.0e38f;
    for (int tile = wid; tile < ntiles; tile += 8) {
        const int m0 = tile << 4;
        for (int r = 0; r < 16; ++r) {
            const int n = m0 + r;
            float4 lo = make_float4(0.f, 0.f, 0.f, 0.f), hi = lo;
            if (n < count) {
                const float4* row = (const float4*)(x + (size_t)(off + n) * D);
                lo = row[lane];
                hi = row[lane + 32];
            }
            *(float4*)(myx + r * XROW + lane * 4)       = lo;
            *(float4*)(myx + r * XROW + 128 + lane * 4) = hi;
        }
        v8f acc = {0.f, 0.f, 0.f, 0.f, 0.f, 0.f, 0.f, 0.f};
        const float* arow = myx + l15 * XROW;
        for (int k = 0; k < D; k += 4) {
            v2f a, bb;
            a.x  = arow[k + kh * 2 + 0];
            a.y  = arow[k + kh * 2 + 1];
            bb.x = s_sw[(k + kh * 2 + 0) * NC + l15];
            bb.y = s_sw[(k + kh * 2 + 1) * NC + l15];
            acc = __builtin_amdgcn_wmma_f32_16x16x4_f32(false, a, false, bb,
                                                        (short)0, acc, false, false);
        }
#pragma unroll
        for (int r = 0; r < 8; ++r) {
            const int n = m0 + r + kh * 8;
            if (n < count) {
                const float sc = acc[r] + sbias;
                scores[(size_t)(off + n) * NC + l15] = sc;
                wmaxv = fmaxf(wmaxv, sc);
            }
        }
    }
    wmaxv = fmaxf(wmaxv, __shfl_xor(wmaxv, 16, 32));
    if (lane < 16) s_red[wid * NC + lane] = wmaxv;
    __threadfence();                  // publish scores block-wide
    __syncthreads();

    float gmax = -3.0e38f;
    for (int w = 0; w < 8; ++w) gmax = fmaxf(gmax, s_red[w * NC + l15]);
    __syncthreads();

    // ---------- Phase 2: sum of exp ----------
    float wsum = 0.f;
    for (int tile = wid; tile < ntiles; tile += 8) {
        const int m0 = tile << 4;
#pragma unroll
        for (int r = 0; r < 8; ++r) {
            const int n = m0 + r + kh * 8;
            if (n < count)
                wsum += __expf(scores[(size_t)(off + n) * NC + l15] - gmax);
        }
    }
    wsum += __shfl_xor(wsum, 16, 32);
    if (lane < 16) s_red[wid * NC + lane] = wsum;
    __syncthreads();
    float gsum = 0.f;
    for (int w = 0; w < 8; ++w) gsum += s_red[w * NC + l15];
    const float rs = 1.0f / gsum;
    __syncthreads();

    // ---------- Phase 3: Y[c, j] = sum_m softmax(m,c) * x[m, j]  (WMMA) ----------
    v8f y0 = {0.f, 0.f, 0.f, 0.f, 0.f, 0.f, 0.f, 0.f};
    v8f y1 = {0.f, 0.f, 0.f, 0.f, 0.f, 0.f, 0.f, 0.f};
    const int c0 = (wid << 5) + l15;     // this wave's column pair
    const int c1 = c0 + 16;
    for (int k0 = 0; k0 < count; k0 += 4) {
        const int n0 = k0 + kh * 2;
        const int n1 = n0 + 1;
        v2f a  = {0.f, 0.f};
        v2f b0 = {0.f, 0.f};
        v2f b1 = {0.f, 0.f};
        if (n0 < count) {
            a.x = __expf(scores[(size_t)(off + n0) * NC + l15] - gmax) * rs;
            const float* row = x + (size_t)(off + n0) * D;
            b0.x = row[c0]; b1.x = row[c1];
        }
        if (n1 < count) {
            a.y = __expf(scores[(size_t)(off + n1) * NC + l15] - gmax) * rs;
            const float* row = x + (size_t)(off + n1) * D;
            b0.y = row[c0]; b1.y = row[c1];
        }
        y0 = __builtin_amdgcn_wmma_f32_16x16x4_f32(false, a, false, b0,
                                                   (short)0, y0, false, false);
        y1 = __builtin_amdgcn_wmma_f32_16x16x4_f32(false, a, false, b1,
                                                   (short)0, y1, false, false);
    }
#pragma unroll
    for (int r = 0; r < 8; ++r) {
        const int c = r + kh * 8;
        s_Y[c * D + c0] = y0[r];
        s_Y[c * D + c1] = y1[r];
    }
    __syncthreads();

    // ---------- Phase 4: epilogue (tiny GEMMs, one output col per thread) ----------
    {   // ctx[s][i] = Y[h*4+s]·Wv_row(i) + bv[i]
        const int i = t, h = i >> 6;
        const float* wrow = in_proj_w + (size_t)(2 * D + i) * D;
        const float bvi = in_proj_b[2 * D + i];
#pragma unroll
        for (int s = 0; s < SEEDS; ++s) {
            const float* yr = s_Y + (h * 4 + s) * D;
            float acc = 0.f;
            for (int j = 0; j < D; ++j) acc += yr[j] * wrow[j];
            s_ctx[s * D + i] = acc + bvi;
        }
    }
    __syncthreads();
    {   // attended[s][o] = ctx[s]·Wout_row(o) + bout[o]
        const int o = t;
        const float* wrow = out_proj_w + (size_t)o * D;
        const float bo = out_proj_b[o];
#pragma unroll
        for (int s = 0; s < SEEDS; ++s) {
            float acc = bo;
            for (int j = 0; j < D; ++j) acc += s_ctx[s * D + j] * wrow[j];
            s_att[s * D + o] = acc;
        }
    }
    __syncthreads();
    {   // h[o] = gelu(flat·fc1_row(o) + b)
        const int o = t;
        const float* wrow = fc1_w + (size_t)o * (SEEDS * D);
        float acc = fc1_b[o];
        for (int u = 0; u < SEEDS * D; ++u) acc += s_att[u] * wrow[u];
        s_h[o] = 0.5f * acc * (1.0f + erff(acc * 0.70710678118654752f));
    }
    __syncthreads();
    {   // out[b][o] = h·fc2_row(o) + b
        const int o = t;
        const float* wrow = fc2_w + (size_t)o * D;
        float acc = fc2_b[o];
        for (int u = 0; u < D; ++u) acc += s_h[u] * wrow[u];
        out[(size_t)b * D + o] = acc;
    }
}

extern "C" void kernel_launch(void* const* d_in, const int* in_sizes, int n_in,
                              void* d_out, int out_size, void* d_ws, size_t ws_size,
                              hipStream_t stream) {
    (void)in_sizes; (void)n_in; (void)out_size; (void)ws_size;
    const float* node_feat  = (const float*)d_in[0];
    const int*   counts     = (const int*)d_in[1];
    const float* seeds      = (const float*)d_in[2];
    const float* in_proj_w  = (const float*)d_in[3];
    const float* in_proj_b  = (const float*)d_in[4];
    const float* out_proj_w = (const float*)d_in[5];
    const float* out_proj_b = (const float*)d_in[6];
    const float* fc1_w      = (const float*)d_in[7];
    const float* fc1_b      = (const float*)d_in[8];
    const float* fc2_w      = (const float*)d_in[9];
    const float* fc2_b      = (const float*)d_in[10];
    float* out = (float*)d_out;
    float* wsf = (float*)d_ws;
    int*   wso = (int*)d_ws;
    float* scores = wsf + WS_SCORES;

    const int dyn_lds = LDS_FLOATS * 4;   // 150016 B (<320KB CDNA5 WGP LDS)
    (void)hipFuncSetAttribute(reinterpret_cast<const void*>(&setx_main),
                              hipFuncAttributeMaxDynamicSharedMemorySize, dyn_lds);

    setx_prep<<<1, 256, 0, stream>>>(counts, seeds, in_proj_w, in_proj_b, wsf, wso);
    setx_main<<<B_GRAPHS, 256, dyn_lds, stream>>>(
        node_feat, wso, wsf, scores,
        in_proj_w, in_proj_b, out_proj_w, out_proj_b,
        fc1_w, fc1_b, fc2_w, fc2_b, out);
}